// GraphConvBlock_5669356832972
// MI455X (gfx1250) — compile-verified
//
#include <hip/hip_runtime.h>

typedef float v2f __attribute__((ext_vector_type(2)));
typedef float v8f __attribute__((ext_vector_type(8)));
typedef int   v4i __attribute__((ext_vector_type(4)));

#define B_    1024
#define V_    64
#define DIN_  256
#define DOUT_ 256
#define EPS_  1e-5f

// ---------------------------------------------------------------------------
// Async global->LDS staging (gfx1250 GLOBAL_LOAD_ASYNC_TO_LDS_B128, ASYNCcnt
// tracked). Builtin params are typed: (global int4*, lds int4*, imm, imm).
// ---------------------------------------------------------------------------
#if defined(__has_builtin)
#if __has_builtin(__builtin_amdgcn_global_load_async_to_lds_b128)
#define HAVE_ASYNC_LDS 1
#endif
#endif

__device__ __forceinline__ void stage_b128(const float* g, float* l) {
#ifdef HAVE_ASYNC_LDS
  __builtin_amdgcn_global_load_async_to_lds_b128(
      (__attribute__((address_space(1))) v4i*)g,
      (__attribute__((address_space(3))) v4i*)l,
      /*offset=*/0, /*cpol=*/0);
#else
  *reinterpret_cast<float4*>(l) = *reinterpret_cast<const float4*>(g);
#endif
}

__device__ __forceinline__ void stage_join() {
#ifdef HAVE_ASYNC_LDS
#if __has_builtin(__builtin_amdgcn_s_wait_asynccnt)
  __builtin_amdgcn_s_wait_asynccnt(0);
#else
  asm volatile("s_wait_asynccnt 0x0" ::: "memory");
#endif
#endif
}

// ---------------------------------------------------------------------------
// Stage 1: h[b,v,o] = feat[b,v,:] @ W[v,:,o] + bias[v,o]
// grid (V, B/32), 256 threads = 8 waves.
// Block owns (vertex v, 32 batch rows); 32x256 feat tile staged in LDS (async).
// Wave w: M-tiles {0,1} x N-tiles {2w, 2w+1}  -> 4 accumulators; each A/B
// fragment feeds 2 WMMAs (halves load traffic per matrix op vs 16-row tiles).
// ---------------------------------------------------------------------------
__global__ __launch_bounds__(256)
void gcb_gemm1(const float* __restrict__ feat, const float* __restrict__ W,
               const float* __restrict__ bias, float* __restrict__ h) {
  __shared__ float ldsA[32 * DIN_];  // 32 KB

  const int v    = blockIdx.x;
  const int bt   = blockIdx.y;       // 32-row batch tile
  const int tid  = threadIdx.x;
  const int lane = tid & 31;
  const int wave = tid >> 5;

  // Cooperative async stage: 32 rows x 256 K = 2048 float4, 8 per thread.
  #pragma unroll
  for (int i = 0; i < 8; ++i) {
    const int e  = tid + i * 256;        // float4 index 0..2047
    const int r  = e >> 6;               // 64 float4 per row
    const int c4 = e & 63;
    stage_b128(feat + ((size_t)(bt * 32 + r) * V_ + v) * DIN_ + (size_t)c4 * 4,
               &ldsA[r * DIN_ + c4 * 4]);
  }
  stage_join();
  __syncthreads();

  const float* Wv = W + (size_t)v * DIN_ * DOUT_;
  const int mr    = lane & 15;
  const int koff  = (lane >> 4) * 2;     // K-pair selector per ISA A-layout
  const int n0    = wave * 32 + (lane & 15);
  const int n1    = n0 + 16;

  v8f c00 = {}, c01 = {}, c10 = {}, c11 = {};

  #pragma unroll 4
  for (int ks = 0; ks < DIN_ / 4; ++ks) {
    const int k = ks * 4 + koff;
    const v2f a0 = *reinterpret_cast<const v2f*>(&ldsA[mr * DIN_ + k]);        // rows 0..15
    const v2f a1 = *reinterpret_cast<const v2f*>(&ldsA[(16 + mr) * DIN_ + k]); // rows 16..31
    v2f b0, b1;
    b0.x = Wv[(size_t)(k + 0) * DOUT_ + n0];
    b0.y = Wv[(size_t)(k + 1) * DOUT_ + n0];
    b1.x = Wv[(size_t)(k + 0) * DOUT_ + n1];
    b1.y = Wv[(size_t)(k + 1) * DOUT_ + n1];
    c00 = __builtin_amdgcn_wmma_f32_16x16x4_f32(false, a0, false, b0, (short)0, c00, false, false);
    c01 = __builtin_amdgcn_wmma_f32_16x16x4_f32(false, a0, false, b1, (short)0, c01, false, false);
    c10 = __builtin_amdgcn_wmma_f32_16x16x4_f32(false, a1, false, b0, (short)0, c10, false, false);
    c11 = __builtin_amdgcn_wmma_f32_16x16x4_f32(false, a1, false, b1, (short)0, c11, false, false);
  }

  // Epilogue: add bias, store. Row i of C lives at m = i + (lane>>4)*8.
  const int rh = (lane >> 4) * 8;
  const float bias0 = bias[v * DOUT_ + n0];
  const float bias1 = bias[v * DOUT_ + n1];
  #pragma unroll
  for (int i = 0; i < 8; ++i) {
    const int r0 = bt * 32 + i + rh;        // M-tile 0
    const int r1 = r0 + 16;                 // M-tile 1
    h[((size_t)r0 * V_ + v) * DOUT_ + n0] = c00[i] + bias0;
    h[((size_t)r0 * V_ + v) * DOUT_ + n1] = c01[i] + bias1;
    h[((size_t)r1 * V_ + v) * DOUT_ + n0] = c10[i] + bias0;
    h[((size_t)r1 * V_ + v) * DOUT_ + n1] = c11[i] + bias1;
  }
}

// ---------------------------------------------------------------------------
// Stage 2a: partial sums / sums-of-squares over 64-row batch chunks.
// grid (V, 16), 256 threads; thread owns one output column o.
// ---------------------------------------------------------------------------
__global__ __launch_bounds__(256)
void gcb_stats_partial(const float* __restrict__ h, float* __restrict__ ps,
                       float* __restrict__ pq) {
  const int v = blockIdx.x;
  const int chunk = blockIdx.y;
  const int o = threadIdx.x;
  float s = 0.f, q = 0.f;
  #pragma unroll 4
  for (int i = 0; i < 64; ++i) {
    const float x = h[((size_t)(chunk * 64 + i) * V_ + v) * DOUT_ + o];
    s += x;
    q += x * x;
  }
  ps[(v * 16 + chunk) * DOUT_ + o] = s;
  pq[(v * 16 + chunk) * DOUT_ + o] = q;
}

// ---------------------------------------------------------------------------
// Stage 2b: scale = gamma*rsqrt(var+eps), shift = beta - mean*scale
// grid V, 256 threads.
// ---------------------------------------------------------------------------
__global__ __launch_bounds__(256)
void gcb_stats_final(const float* __restrict__ ps, const float* __restrict__ pq,
                     const float* __restrict__ gamma, const float* __restrict__ beta,
                     float* __restrict__ scale, float* __restrict__ shift) {
  const int v = blockIdx.x;
  const int o = threadIdx.x;
  float s = 0.f, q = 0.f;
  #pragma unroll
  for (int c = 0; c < 16; ++c) {
    s += ps[(v * 16 + c) * DOUT_ + o];
    q += pq[(v * 16 + c) * DOUT_ + o];
  }
  const float mean = s * (1.0f / B_);
  const float var  = q * (1.0f / B_) - mean * mean;
  const float sc   = gamma[v * DOUT_ + o] * rsqrtf(var + EPS_);
  scale[v * DOUT_ + o] = sc;
  shift[v * DOUT_ + o] = beta[v * DOUT_ + o] - mean * sc;
}

// ---------------------------------------------------------------------------
// Stage 3: out[b] = relu(adj @ h_bn[b])   [64x64] x [64x256]
// grid B, 256 threads = 8 waves. adj staged async; h_bn staged through VGPRs
// (BN applied en route). Wave w: M-tile w%4, N-tiles (w/4)*8 .. +7.
// ---------------------------------------------------------------------------
__global__ __launch_bounds__(256)
void gcb_mix(const float* __restrict__ h, const float* __restrict__ adj,
             const float* __restrict__ scale, const float* __restrict__ shift,
             float* __restrict__ out) {
  __shared__ float ldsH[V_ * DOUT_];  // 64 KB
  __shared__ float ldsAdj[V_ * V_];   // 16 KB

  const int b    = blockIdx.x;
  const int tid  = threadIdx.x;
  const int lane = tid & 31;
  const int wave = tid >> 5;

  // adj: 1024 float4, 4 per thread, async global->LDS.
  #pragma unroll
  for (int i = 0; i < 4; ++i) {
    const int e = tid + i * 256;       // float4 index 0..1023
    stage_b128(adj + (size_t)e * 4, &ldsAdj[e * 4]);
  }
  // h_bn[b][v][o] = h[b,v,o]*scale[v,o] + shift[v,o]; flat layouts line up.
  #pragma unroll 8
  for (int i = 0; i < 64; ++i) {
    const int idx = tid + i * 256;     // 0..16383 over [v][o]
    ldsH[idx] = h[(size_t)b * V_ * DOUT_ + idx] * scale[idx] + shift[idx];
  }
  stage_join();
  __syncthreads();

  const int mt     = wave & 3;
  const int ntBase = (wave >> 2) * 8;
  const int mr     = lane & 15;
  const int koff   = (lane >> 4) * 2;
  const int n      = lane & 15;

  v8f acc[8] = {};

  #pragma unroll
  for (int ks = 0; ks < V_ / 4; ++ks) {
    const int k = ks * 4 + koff;
    const v2f a = *reinterpret_cast<const v2f*>(&ldsAdj[(mt * 16 + mr) * V_ + k]);
    #pragma unroll
    for (int t = 0; t < 8; ++t) {
      const int col = (ntBase + t) * 16 + n;
      v2f bf;
      bf.x = ldsH[(k + 0) * DOUT_ + col];
      bf.y = ldsH[(k + 1) * DOUT_ + col];
      acc[t] = __builtin_amdgcn_wmma_f32_16x16x4_f32(false, a, false, bf, (short)0, acc[t], false, false);
    }
  }

  const int rh = (lane >> 4) * 8;
  #pragma unroll
  for (int t = 0; t < 8; ++t) {
    const int col = (ntBase + t) * 16 + n;
    #pragma unroll
    for (int i = 0; i < 8; ++i) {
      const int u = mt * 16 + i + rh;
      const float x = acc[t][i];
      out[((size_t)b * V_ + u) * DOUT_ + col] = x > 0.f ? x : 0.f;
    }
  }
}

// ---------------------------------------------------------------------------
extern "C" void kernel_launch(void* const* d_in, const int* in_sizes, int n_in,
                              void* d_out, int out_size, void* d_ws, size_t ws_size,
                              hipStream_t stream) {
  const float* feat  = (const float*)d_in[0];  // [B,V,DIN]
  const float* adj   = (const float*)d_in[1];  // [V,V]
  const float* W     = (const float*)d_in[2];  // [V,DIN,DOUT]
  const float* bias  = (const float*)d_in[3];  // [V,DOUT]
  const float* gamma = (const float*)d_in[4];  // [V,DOUT]
  const float* beta  = (const float*)d_in[5];  // [V,DOUT]
  float* out = (float*)d_out;                  // [B,V,DOUT]

  float* ws    = (float*)d_ws;
  float* h     = ws;                                   // B*V*DOUT
  float* ps    = h  + (size_t)B_ * V_ * DOUT_;         // V*16*DOUT
  float* pq    = ps + (size_t)V_ * 16 * DOUT_;         // V*16*DOUT
  float* scale = pq + (size_t)V_ * 16 * DOUT_;         // V*DOUT
  float* shift = scale + (size_t)V_ * DOUT_;           // V*DOUT

  gcb_gemm1<<<dim3(V_, B_ / 32), 256, 0, stream>>>(feat, W, bias, h);
  gcb_stats_partial<<<dim3(V_, 16), 256, 0, stream>>>(h, ps, pq);
  gcb_stats_final<<<V_, 256, 0, stream>>>(ps, pq, gamma, beta, scale, shift);
  gcb_mix<<<B_, 256, 0, stream>>>(h, adj, scale, shift, out);
}